// GCN_10531259810641
// MI455X (gfx1250) — compile-verified
//
#include <hip/hip_runtime.h>

typedef __attribute__((ext_vector_type(2))) float v2f;
typedef __attribute__((ext_vector_type(8))) float v8f;
typedef __attribute__((ext_vector_type(4))) unsigned int v4u;
typedef __attribute__((ext_vector_type(8))) int v8i;
typedef __attribute__((ext_vector_type(4))) int v4i;

#define IN_C 128
#define HID 128
#define OUT_C 64

// ---------------------------------------------------------------- utilities

__global__ void fill_kernel(float* __restrict__ p, float v, long long n) {
    long long i = (long long)blockIdx.x * blockDim.x + threadIdx.x;
    if (i < n) p[i] = v;
}

__global__ void degree_kernel(const long long* __restrict__ cols,
                              float* __restrict__ deg, int E) {
    int i = blockIdx.x * blockDim.x + threadIdx.x;
    if (i < E) atomicAdd(&deg[(int)cols[i]], 1.0f);
}

__global__ void dinv_kernel(const float* __restrict__ deg,
                            float* __restrict__ dinv, int n) {
    int i = blockIdx.x * blockDim.x + threadIdx.x;
    if (i < n) {
        float d = deg[i];
        dinv[i] = (d > 0.0f) ? rsqrtf(d) : 0.0f;
    }
}

__global__ void bias_relu_kernel(float* __restrict__ h,
                                 const float* __restrict__ b, long long total) {
    long long i = (long long)blockIdx.x * blockDim.x + threadIdx.x;
    if (i < total) {
        float v = h[i] + b[(int)(i & (HID - 1))];
        h[i] = fmaxf(v, 0.0f);
    }
}

// ------------------------------------------------- WMMA fp32 GEMM (+bias/relu)
// C[nrows, NC] = A[nrows, K] @ W[K, NC]   (FLAGS: 1 = +bias, 2 = relu)
// Block = 128 threads = 4 waves; each wave computes a 16 x NC tile.
// W ([K,NC] f32, <= 64 KB) is staged into LDS by the Tensor Data Mover:
// wave 0 issues one tensor_load_to_lds D# (2D tile: dim0=NC, dim1=K,
// stride0=NC, data_size=4B, LDS offset 0 = sW), waits TENSORcnt, barrier.
//
// V_WMMA_F32_16X16X4_F32 operand layout (ISA 7.12.2):
//   A (16x4): lanes 0-15 hold M=lane, VGPR0=K0, VGPR1=K1; lanes 16-31: K2,K3.
//   B (4x16): row-striped mirroring C: VGPR j holds row j (lanes 0-15) and
//             row j+2 (lanes 16-31), N = lane&15.
//   C (16x16): VGPR v holds M=v (lanes 0-15) and M=v+8 (lanes 16-31).
template <int K, int NC, int FLAGS>
__global__ __launch_bounds__(128) void wmma_gemm_kernel(
    const float* __restrict__ A, const float* __restrict__ W,
    const float* __restrict__ bias, float* __restrict__ C,
    int nrows) {
    __shared__ float sW[K * NC];

#if __has_builtin(__builtin_amdgcn_tensor_load_to_lds)
    if (threadIdx.x < 32) {   // wave 0 only: TDM ignores EXEC, issue once
        const unsigned long long ga = (unsigned long long)(const void*)W;
        // D# group 0: count=1, lds_addr=0 (sW is at LDS offset 0),
        // global_addr[56:0] at bits [120:64], type=2 at bits [127:126].
        v4u g0 = { 1u, 0u, (unsigned)ga,
                   ((unsigned)(ga >> 32) & 0x01FFFFFFu) | 0x80000000u };
        // D# group 1: data_size=2 (4B) at [17:16]; tensor_dim0=NC @ [79:48];
        // tensor_dim1=K @ [111:80]; tile_dim0=NC @ [127:112];
        // tile_dim1=K @ [143:128]; tensor_dim0_stride=NC @ [207:160].
        v8i g1 = { 0x00020000, NC << 16, K << 16, NC << 16, K, NC, 0, 0 };
        v4i gz = { 0, 0, 0, 0 };
#if __clang_major__ >= 23
        v8i gz8 = { 0, 0, 0, 0, 0, 0, 0, 0 };
        __builtin_amdgcn_tensor_load_to_lds(g0, g1, gz, gz, gz8, 0);
#else
        __builtin_amdgcn_tensor_load_to_lds(g0, g1, gz, gz, 0);
#endif
        __builtin_amdgcn_s_wait_tensorcnt(0);
    }
    __syncthreads();
#else
    for (int i = threadIdx.x * 4; i < K * NC; i += blockDim.x * 4) {
        *(float4*)(sW + i) = *(const float4*)(W + i);
    }
    __syncthreads();
#endif

    const int lane = threadIdx.x & 31;
    const int wave = threadIdx.x >> 5;
    const int m    = lane & 15;     // M index for A / N index for B,C
    const int kg   = lane >> 4;     // K-half select (0: K0/K1, 1: K2/K3)
    const int r0   = (blockIdx.x * 4 + wave) * 16;

    int rowc = r0 + m;
    if (rowc > nrows - 1) rowc = nrows - 1;   // clamp loads; EXEC stays all-1
    const float* arow = A + (size_t)rowc * K;

    constexpr int NT = NC / 16;
    v8f acc[NT];
#pragma unroll
    for (int t = 0; t < NT; ++t) {
        v8f z = {0.f, 0.f, 0.f, 0.f, 0.f, 0.f, 0.f, 0.f};
        acc[t] = z;
    }

    for (int k = 0; k < K; k += 4) {
        v2f a;
        a.x = arow[k + kg * 2 + 0];
        a.y = arow[k + kg * 2 + 1];
#pragma unroll
        for (int t = 0; t < NT; ++t) {
            const int ncol = t * 16 + m;
            v2f b;
            b.x = sW[(k + kg * 2 + 0) * NC + ncol];
            b.y = sW[(k + kg * 2 + 1) * NC + ncol];
            acc[t] = __builtin_amdgcn_wmma_f32_16x16x4_f32(
                false, a, false, b, (short)0, acc[t], false, false);
        }
    }

#pragma unroll
    for (int t = 0; t < NT; ++t) {
        const int col = t * 16 + m;
        const float bv = (FLAGS & 1) ? bias[col] : 0.0f;   // hoisted per tile
#pragma unroll
        for (int v = 0; v < 8; ++v) {
            const int row = r0 + v + kg * 8;
            if (row < nrows) {
                float val = acc[t][v] + bv;
                if (FLAGS & 2) val = fmaxf(val, 0.0f);
                C[(size_t)row * NC + col] = val;
            }
        }
    }
}

// ------------------------------------------------------- edge scatter (GCN msg)
// out[col] += h[row] * dinv[row] * dinv[col]  for E edges + N self loops.
// One wave per edge; each lane owns 4 of the 128 channels.
__global__ __launch_bounds__(256) void scatter_kernel(
    const float* __restrict__ h, const long long* __restrict__ ei,
    const float* __restrict__ dinv, float* __restrict__ out,
    int E, int n) {
    const int lane = threadIdx.x & 31;
    const int wv   = threadIdx.x >> 5;
    const long long idx   = (long long)blockIdx.x * 8 + wv;
    const long long total = (long long)E + n;
    if (idx >= total) return;

    int r, c;
    if (idx < E) {
        // speculative prefetch of the edge stream ahead (global_prefetch_b8)
        __builtin_prefetch(&ei[idx + 4096], 0, 0);
        __builtin_prefetch(&ei[E + idx + 4096], 0, 0);
        r = (int)ei[idx];
        c = (int)ei[E + idx];
    } else {
        r = c = (int)(idx - E);   // self loop
    }

    const float nrm = dinv[r] * dinv[c];
    const float4 v  = ((const float4*)(h + (size_t)r * HID))[lane];
    float* orow     = out + (size_t)c * HID + lane * 4;
    atomicAdd(orow + 0, v.x * nrm);
    atomicAdd(orow + 1, v.y * nrm);
    atomicAdd(orow + 2, v.z * nrm);
    atomicAdd(orow + 3, v.w * nrm);
}

// -------------------------------------------------------------------- driver

extern "C" void kernel_launch(void* const* d_in, const int* in_sizes, int n_in,
                              void* d_out, int out_size, void* d_ws, size_t ws_size,
                              hipStream_t stream) {
    const float*     x      = (const float*)d_in[0];
    const long long* ei     = (const long long*)d_in[1];   // [2, E] int64
    const float*     W_in   = (const float*)d_in[2];
    const float*     b_in   = (const float*)d_in[3];
    const float*     conv_W = (const float*)d_in[4];       // [3, HID, HID]
    const float*     conv_b = (const float*)d_in[5];       // [3, HID]
    const float*     W_out  = (const float*)d_in[6];
    const float*     b_out  = (const float*)d_in[7];
    float*           out    = (float*)d_out;

    const int N = in_sizes[0] / IN_C;
    const int E = in_sizes[1] / 2;

    float* deg  = (float*)d_ws;
    float* dinv = deg + N;
    float* hA   = dinv + N;
    float* hB   = hA + (size_t)N * HID;

    const long long NH = (long long)N * HID;

    // degrees (self loop contributes 1) and symmetric-norm rsqrt
    fill_kernel<<<(N + 255) / 256, 256, 0, stream>>>(deg, 1.0f, N);
    degree_kernel<<<(E + 255) / 256, 256, 0, stream>>>(ei + E, deg, E);
    dinv_kernel<<<(N + 255) / 256, 256, 0, stream>>>(deg, dinv, N);

    const int gemm_grid = (N + 63) / 64;

    // hA = relu(x @ W_in + b_in)
    wmma_gemm_kernel<IN_C, HID, 3><<<gemm_grid, 128, 0, stream>>>(
        x, W_in, b_in, hA, N);

    const long long totalEdges = (long long)E + N;
    const int scat_grid = (int)((totalEdges + 7) / 8);

    for (int l = 0; l < 3; ++l) {
        // hB = hA @ conv_W[l]
        wmma_gemm_kernel<HID, HID, 0><<<gemm_grid, 128, 0, stream>>>(
            hA, conv_W + (size_t)l * HID * HID, conv_b, hB, N);
        // hA = segment_sum(hB[rows] * norm, cols)
        fill_kernel<<<(int)((NH + 255) / 256), 256, 0, stream>>>(hA, 0.0f, NH);
        scatter_kernel<<<scat_grid, 256, 0, stream>>>(hB, ei, dinv, hA, E, N);
        // hA = relu(hA + conv_b[l])
        bias_relu_kernel<<<(int)((NH + 255) / 256), 256, 0, stream>>>(
            hA, conv_b + (size_t)l * HID, NH);
    }

    // out = hA @ W_out + b_out
    wmma_gemm_kernel<HID, OUT_C, 1><<<gemm_grid, 128, 0, stream>>>(
        hA, W_out, b_out, out, N);

    (void)n_in; (void)out_size; (void)ws_size;
}